// SkinningWarp_79182017069768
// MI455X (gfx1250) — compile-verified
//
#include <hip/hip_runtime.h>

typedef __attribute__((ext_vector_type(2))) float v2f;
typedef __attribute__((ext_vector_type(8))) float v8f;

#define MB 25   // bones
#define NP 64   // points per m  (N*D = 64*1)

__device__ __forceinline__ void qmul4(const float a[4], const float b[4], float o[4]) {
    // (aw,av)*(bw,bv): w = aw*bw - av.bv ; v = aw*bv + bw*av + av x bv
    o[0] = a[0]*b[0] - a[1]*b[1] - a[2]*b[2] - a[3]*b[3];
    o[1] = a[0]*b[1] + b[0]*a[1] + (a[2]*b[3] - a[3]*b[2]);
    o[2] = a[0]*b[2] + b[0]*a[2] + (a[3]*b[1] - a[1]*b[3]);
    o[3] = a[0]*b[3] + b[0]*a[3] + (a[1]*b[2] - a[2]*b[1]);
}

__global__ __launch_bounds__(128)
void dqskin_wmma_kernel(const float* __restrict__ xyz,
                        const float* __restrict__ t_qr,
                        const float* __restrict__ t_qd,
                        const float* __restrict__ rest_qr,
                        const float* __restrict__ rest_qd,
                        const float* __restrict__ log_gauss,
                        float* __restrict__ out)
{
    __shared__ float sQR[MB][4];     // se3 qr before sign fix
    __shared__ float qrS[MB][4];     // sign-fixed se3 qr
    __shared__ float qdS[MB][4];     // sign-fixed se3 qd
    __shared__ float Wl[32][12];     // quadratic-form coefficients per bone (padded)
    __shared__ float dl[NP][32];     // dist2[point][bone]

    const int m   = blockIdx.x;
    const int tid = threadIdx.x;

    float se3r[4] = {0.f,0.f,0.f,0.f};
    float se3d[4] = {0.f,0.f,0.f,0.f};

    // ---------------- Phase 1: per-bone setup (25 lanes) ----------------
    if (tid < MB) {
        const int b = tid;
        const float* tq = t_qr   + (m*MB + b)*4;
        const float* td = t_qd   + (m*MB + b)*4;
        const float* rq = rest_qr+ (m*MB + b)*4;
        const float* rd = rest_qd+ (m*MB + b)*4;
        float tqv[4] = {tq[0],  tq[1],  tq[2],  tq[3]};
        float tdv[4] = {td[0],  td[1],  td[2],  td[3]};
        float rqv[4] = {rq[0],  rq[1],  rq[2],  rq[3]};
        float rqc[4] = {rq[0], -rq[1], -rq[2], -rq[3]};
        float rdc[4] = {rd[0], -rd[1], -rd[2], -rd[3]};

        // se3 = t_art * inverse(rest_art)
        qmul4(tqv, rqc, se3r);
        float u1[4], u2[4];
        qmul4(tqv, rdc, u1);
        qmul4(tdv, rqc, u2);
        #pragma unroll
        for (int k = 0; k < 4; ++k) se3d[k] = u1[k] + u2[k];
        #pragma unroll
        for (int k = 0; k < 4; ++k) sQR[b][k] = se3r[k];

        // inverse-rest affine transform: x_bone = R x + t, R = rot(conj(rest_qr))
        const float w0 = rqc[0], ax = rqc[1], ay = rqc[2], az = rqc[3];
        const float R00 = 1.f - 2.f*(ay*ay + az*az);
        const float R01 = 2.f*(ax*ay - w0*az);
        const float R02 = 2.f*(ax*az + w0*ay);
        const float R10 = 2.f*(ax*ay + w0*az);
        const float R11 = 1.f - 2.f*(ax*ax + az*az);
        const float R12 = 2.f*(ay*az - w0*ax);
        const float R20 = 2.f*(ax*az - w0*ay);
        const float R21 = 2.f*(ay*az + w0*ax);
        const float R22 = 1.f - 2.f*(ax*ax + ay*ay);
        float tt[4];
        qmul4(rdc, rqv, tt);                       // t = 2*(qd* x conj(qd*->qr))
        const float tx = 2.f*tt[1], ty = 2.f*tt[2], tz = 2.f*tt[3];

        const float i0 = __expf(-log_gauss[b*3+0]);   // 1/scale
        const float i1 = __expf(-log_gauss[b*3+1]);
        const float i2 = __expf(-log_gauss[b*3+2]);
        const float A00=R00*i0, A01=R01*i0, A02=R02*i0;
        const float A10=R10*i1, A11=R11*i1, A12=R12*i1;
        const float A20=R20*i2, A21=R21*i2, A22=R22*i2;
        const float c0=tx*i0, c1=ty*i1, c2=tz*i2;

        // dist2 = f(x) . w ; f = [x^2,y^2,z^2, xy,xz,yz, x,y,z, 1, 0,0]
        Wl[b][0]  = A00*A00 + A10*A10 + A20*A20;
        Wl[b][1]  = A01*A01 + A11*A11 + A21*A21;
        Wl[b][2]  = A02*A02 + A12*A12 + A22*A22;
        Wl[b][3]  = 2.f*(A00*A01 + A10*A11 + A20*A21);
        Wl[b][4]  = 2.f*(A00*A02 + A10*A12 + A20*A22);
        Wl[b][5]  = 2.f*(A01*A02 + A11*A12 + A21*A22);
        Wl[b][6]  = 2.f*(A00*c0 + A10*c1 + A20*c2);
        Wl[b][7]  = 2.f*(A01*c0 + A11*c1 + A21*c2);
        Wl[b][8]  = 2.f*(A02*c0 + A12*c1 + A22*c2);
        Wl[b][9]  = c0*c0 + c1*c1 + c2*c2;
        Wl[b][10] = 0.f;
        Wl[b][11] = 0.f;
    } else if (tid < 32) {
        #pragma unroll
        for (int k = 0; k < 12; ++k) Wl[tid][k] = 0.f;   // pad bones 25..31
    }
    __syncthreads();

    // sign fix vs bone 0 (matches sign(dot)==0 -> +1)
    if (tid < MB) {
        const float d = se3r[0]*sQR[0][0] + se3r[1]*sQR[0][1]
                      + se3r[2]*sQR[0][2] + se3r[3]*sQR[0][3];
        const float sgn = (d >= 0.f) ? 1.f : -1.f;
        #pragma unroll
        for (int k = 0; k < 4; ++k) { qrS[tid][k] = sgn*se3r[k]; qdS[tid][k] = sgn*se3d[k]; }
    }
    __syncthreads();

    // ---------------- Phase 2: dist2 via V_WMMA_F32_16X16X4_F32 ----------------
    {
        const int ln  = tid & 31;
        const int wv  = tid >> 5;          // wave -> 16-point tile
        const bool hiL = (ln >= 16);
        const int row = ln & 15;

        const float* xp = xyz + ((size_t)m*NP + wv*16 + row)*3;
        const float x = xp[0], y = xp[1], z = xp[2];

        // A matrix 16x4 f32 layout: VGPR0 = K{4s}(lo)/K{4s+2}(hi), VGPR1 = K{4s+1}/K{4s+3}
        v2f a0, a1, a2;
        a0[0] = hiL ? z*z : x*x;   a0[1] = hiL ? x*y : y*y;    // K 0..3
        a1[0] = hiL ? x   : x*z;   a1[1] = hiL ? y   : y*z;    // K 4..7
        a2[0] = hiL ? 0.f : z;     a2[1] = hiL ? 0.f : 1.f;    // K 8..11
        const int ko = hiL ? 2 : 0;

        #pragma unroll
        for (int t = 0; t < 2; ++t) {
            const int bone = t*16 + row;                  // B col = bone
            const float* wr = &Wl[bone][0];
            v2f b0, b1, b2;
            b0[0] = wr[0+ko]; b0[1] = wr[1+ko];
            b1[0] = wr[4+ko]; b1[1] = wr[5+ko];
            b2[0] = wr[8+ko]; b2[1] = wr[9+ko];
            v8f c = {0.f,0.f,0.f,0.f,0.f,0.f,0.f,0.f};
            c = __builtin_amdgcn_wmma_f32_16x16x4_f32(false, a0, false, b0, (short)0, c, false, false);
            c = __builtin_amdgcn_wmma_f32_16x16x4_f32(false, a1, false, b1, (short)0, c, false, false);
            c = __builtin_amdgcn_wmma_f32_16x16x4_f32(false, a2, false, b2, (short)0, c, false, false);
            const int pb = wv*16 + (hiL ? 8 : 0);         // D rows: M=r (lo) / r+8 (hi)
            #pragma unroll
            for (int r = 0; r < 8; ++r) dl[pb + r][bone] = c[r];
        }
    }
    __syncthreads();

    // ---------------- Phase 3: exp-blend + apply (one point per lane) ----------------
    if (tid < NP) {
        const int p = tid;
        float dmin = 3.4e38f;
        #pragma unroll
        for (int b = 0; b < MB; ++b) dmin = fminf(dmin, dl[p][b]);

        // softmax denominator cancels under the final |qr| renormalization
        float q0=0.f,q1=0.f,q2=0.f,q3=0.f, d0=0.f,d1=0.f,d2=0.f,d3=0.f;
        #pragma unroll
        for (int b = 0; b < MB; ++b) {
            const float e = __expf(0.5f*(dmin - dl[p][b]));
            q0 += e*qrS[b][0]; q1 += e*qrS[b][1]; q2 += e*qrS[b][2]; q3 += e*qrS[b][3];
            d0 += e*qdS[b][0]; d1 += e*qdS[b][1]; d2 += e*qdS[b][2]; d3 += e*qdS[b][3];
        }
        const float n2  = q0*q0 + q1*q1 + q2*q2 + q3*q3;
        const float rin = rsqrtf(n2);
        const float w  = q0*rin, vx = q1*rin, vy = q2*rin, vz = q3*rin;

        const float* xp = xyz + ((size_t)m*NP + p)*3;
        const float x = xp[0], y = xp[1], z = xp[2];
        // qrotate(qn, x)
        const float ux = vy*z - vz*y, uy = vz*x - vx*z, uz = vx*y - vy*x;
        const float rx = x + 2.f*(w*ux + (vy*uz - vz*uy));
        const float ry = y + 2.f*(w*uy + (vz*ux - vx*uz));
        const float rz = z + 2.f*(w*uz + (vx*uy - vy*ux));
        // translation: 2*rin^2 * vec(qmul(qd_u, conj(qr_u)))
        const float s2 = 2.f*rin*rin;
        const float tx = s2*(-d0*q1 + q0*d1 - d2*q3 + d3*q2);
        const float ty = s2*(-d0*q2 + q0*d2 - d3*q1 + d1*q3);
        const float tz = s2*(-d0*q3 + q0*d3 - d1*q2 + d2*q1);

        float* op = out + ((size_t)m*NP + p)*3;
        op[0] = rx + tx; op[1] = ry + ty; op[2] = rz + tz;
    }
}

extern "C" void kernel_launch(void* const* d_in, const int* in_sizes, int n_in,
                              void* d_out, int out_size, void* d_ws, size_t ws_size,
                              hipStream_t stream) {
    (void)n_in; (void)out_size; (void)d_ws; (void)ws_size;
    const float* xyz     = (const float*)d_in[0];
    const float* t_qr    = (const float*)d_in[1];
    const float* t_qd    = (const float*)d_in[2];
    const float* rest_qr = (const float*)d_in[3];
    const float* rest_qd = (const float*)d_in[4];
    const float* lg      = (const float*)d_in[5];
    float* out = (float*)d_out;

    const int M = in_sizes[1] / (MB * 4);   // t_qr has M*25*4 elements
    dqskin_wmma_kernel<<<dim3(M), dim3(128), 0, stream>>>(
        xyz, t_qr, t_qd, rest_qr, rest_qd, lg, out);
}